// LuongAttnDecoderRNN_74242804679159
// MI455X (gfx1250) — compile-verified
//
#include <hip/hip_runtime.h>
#include <hip/hip_bf16.h>

// S=2048, B=64, H=512, E=3
// energy[b,s] = <hidden[b,:], EO[s,b,:]> + (hidden[b,:] @ A) . emb[s,b,:]
// out[b,0,s] = softmax_s(energy[b,s])
//
// Memory-bound: 268MB one-shot stream of encoder_outputs @ 23.3 TB/s ~= 11.6us floor.
// Use V_WMMA_F32_16X16X4_F32 (native fp32 WMMA) for the batched GEMV:
//   A = 16 seq-rows x 4 h-elems of EO, B = hidden chunk broadcast to all 16 cols.
// EO loads are non-temporal (each byte read exactly once; working set > 192MB L2).

typedef __attribute__((ext_vector_type(2))) float v2f;
typedef __attribute__((ext_vector_type(4))) float v4f;
typedef __attribute__((ext_vector_type(8))) float v8f;

__global__ __launch_bounds__(32) void energy_wmma_kernel(
    const float* __restrict__ hidden,   // [B,H]
    const float* __restrict__ enc,      // [S,B,H]
    const float* __restrict__ emb,      // [S,B,E]
    const float* __restrict__ affect,   // [H,E]
    float* __restrict__ energy)         // [B,S]  (== d_out, softmaxed later)
{
    constexpr int B = 64, H = 512, S = 2048;

    const int lane = threadIdx.x;       // 0..31 (wave32)
    const int b    = blockIdx.x;        // batch
    const int s0   = blockIdx.y << 4;   // 16-row seq tile
    const int m    = lane & 15;         // seq row within tile (A-matrix M)
    const int half = lane >> 4;         // 0: K-slots {0,1}, 1: K-slots {2,3}

    // A-matrix ISA layout (32-bit A 16x4): lanes 0-15 hold K=0,1 ; lanes 16-31 hold K=2,3.
    // We assign K-slot->h mapping: half 0 covers h+0..h+3 over two WMMAs, half 1 covers h+4..h+7.
    const float* aPtr = enc    + ((size_t)(s0 + m) * B + b) * H + (half << 2);
    const float* hPtr = hidden + (size_t)b * H + (half << 2);

    v8f acc = {};

    #pragma unroll 4
    for (int k = 0; k < H; k += 8) {
        // streamed, read-once -> non-temporal; 16B/lane, rows advance sequentially
        v4f a4 = __builtin_nontemporal_load((const v4f*)(aPtr + k));
        v4f h4 = *(const v4f*)(hPtr + k);    // broadcast within half-wave, L0/L2 resident

        v2f aLo = {a4.x, a4.y};
        v2f bLo = {h4.x, h4.y};              // B rows K0/K1 (lanes<16) = h+0,h+1 ; K2/K3 = h+4,h+5
        acc = __builtin_amdgcn_wmma_f32_16x16x4_f32(
            false, aLo, false, bLo, (short)0, acc, false, false);

        v2f aHi = {a4.z, a4.w};
        v2f bHi = {h4.z, h4.w};              // K0/K1 = h+2,h+3 ; K2/K3 = h+6,h+7
        acc = __builtin_amdgcn_wmma_f32_16x16x4_f32(
            false, aHi, false, bHi, (short)0, acc, false, false);
    }

    // ---- affect term: hA[e] = sum_h hidden[b,h]*A[h,e]  (E=3, tiny) ----
    const float* hb = hidden + (size_t)b * H;
    float p0 = 0.f, p1 = 0.f, p2 = 0.f;
    for (int h = lane; h < H; h += 32) {
        const float hv = hb[h];
        p0 = fmaf(hv, affect[h * 3 + 0], p0);
        p1 = fmaf(hv, affect[h * 3 + 1], p1);
        p2 = fmaf(hv, affect[h * 3 + 2], p2);
    }
    #pragma unroll
    for (int off = 16; off > 0; off >>= 1) {
        p0 += __shfl_xor(p0, off, 32);
        p1 += __shfl_xor(p1, off, 32);
        p2 += __shfl_xor(p2, off, 32);
    }
    // lane L computes affect energy for seq row m = L&15
    const float* em = emb + ((size_t)(s0 + m) * B + b) * 3;
    const float aff = fmaf(p0, em[0], fmaf(p1, em[1], p2 * em[2]));

    // D layout: VGPR v holds M=v (lanes 0-15) / M=v+8 (lanes 16-31); column N=0 on lanes 0,16.
    // Lane 0 owns rows 0..7, lane 16 owns rows 8..15. Gather matching affect values.
    float affv[8];
    #pragma unroll
    for (int j = 0; j < 8; ++j) affv[j] = __shfl(aff, (half << 3) + j, 32);

    if (m == 0) {  // lanes 0 and 16
        float* op = energy + (size_t)b * S + s0 + (half << 3);
        v4f o0 = {acc[0] + affv[0], acc[1] + affv[1], acc[2] + affv[2], acc[3] + affv[3]};
        v4f o1 = {acc[4] + affv[4], acc[5] + affv[5], acc[6] + affv[6], acc[7] + affv[7]};
        *(v4f*)(op)     = o0;
        *(v4f*)(op + 4) = o1;
    }
}

// In-place row softmax over S=2048, one block per batch row, 256 threads (8 waves).
__global__ __launch_bounds__(256) void softmax_rows_kernel(float* __restrict__ e)
{
    constexpr int S = 2048;
    float* row = e + (size_t)blockIdx.x * S;
    const int tid  = threadIdx.x;
    const int wave = tid >> 5;

    float v[8];
    float mx = -3.402823466e+38f;
    #pragma unroll
    for (int j = 0; j < 8; ++j) {
        v[j] = row[tid + 256 * j];
        mx = fmaxf(mx, v[j]);
    }
    #pragma unroll
    for (int off = 16; off > 0; off >>= 1) mx = fmaxf(mx, __shfl_xor(mx, off, 32));

    __shared__ float red[8];
    if ((tid & 31) == 0) red[wave] = mx;
    __syncthreads();
    float bmax = red[0];
    #pragma unroll
    for (int i = 1; i < 8; ++i) bmax = fmaxf(bmax, red[i]);

    float s = 0.f;
    #pragma unroll
    for (int j = 0; j < 8; ++j) {
        v[j] = __expf(v[j] - bmax);
        s += v[j];
    }
    #pragma unroll
    for (int off = 16; off > 0; off >>= 1) s += __shfl_xor(s, off, 32);

    __syncthreads();               // done reading red (max) before reuse
    if ((tid & 31) == 0) red[wave] = s;
    __syncthreads();
    float bsum = 0.f;
    #pragma unroll
    for (int i = 0; i < 8; ++i) bsum += red[i];

    const float inv = 1.0f / bsum;
    #pragma unroll
    for (int j = 0; j < 8; ++j) row[tid + 256 * j] = v[j] * inv;
}

extern "C" void kernel_launch(void* const* d_in, const int* in_sizes, int n_in,
                              void* d_out, int out_size, void* d_ws, size_t ws_size,
                              hipStream_t stream) {
    const float* hidden = (const float*)d_in[0];   // [1,64,512]
    const float* enc    = (const float*)d_in[1];   // [2048,64,512]
    const float* emb    = (const float*)d_in[2];   // [2048,64,3]
    const float* affect = (const float*)d_in[3];   // [512,3]
    float* out = (float*)d_out;                    // [64,1,2048]

    dim3 gridA(64, 2048 / 16);                     // (batch, seq-tile)
    energy_wmma_kernel<<<gridA, 32, 0, stream>>>(hidden, enc, emb, affect, out);
    softmax_rows_kernel<<<64, 256, 0, stream>>>(out);
}